// CrossAttentionFusion_59150289600562
// MI455X (gfx1250) — compile-verified
//
#include <hip/hip_runtime.h>

// ---------------------------------------------------------------------------
// CDNA5 (gfx1250) fused cross-attention kernel set.
//   * Five fp32 GEMMs on the matrix pipe: V_WMMA_F32_16X16X4_F32 (exact fp32)
//   * Global->LDS staging via GLOBAL_LOAD_ASYNC_TO_LDS_B128 (ASYNCcnt),
//     double-buffered so memory fully overlaps the WMMA stream
//   * Two row-wise LayerNorm kernels (memory-trivial at 23.3 TB/s)
// ---------------------------------------------------------------------------

typedef __attribute__((ext_vector_type(2))) float v2f;
typedef __attribute__((ext_vector_type(8))) float v8f;

#define LDSS 36                       // padded LDS row stride (floats)
#define TILE_FLOATS (128 * LDSS)      // one 128x32 tile buffer (padded)
#define TILE_BYTES  (TILE_FLOATS * 4)

__device__ __forceinline__ void async_b128_to_lds(unsigned ldsAddr,
                                                  int gByteOff,
                                                  const float* sbase) {
    // GVS mode: mem = SGPR64 + VGPR32 + inst_offset ; LDS dst addr in VGPR
    asm volatile("global_load_async_to_lds_b128 %0, %1, %2"
                 :: "v"(ldsAddr), "v"(gByteOff), "s"(sbase)
                 : "memory");
}

// out[M=gridDim.y*128, N=1024] = A[.,K] * W[1024,K]^T + bias (+ residual)
__global__ __launch_bounds__(256) void gemm_f32_wmma(
    const float* __restrict__ A, int lda,
    const float* __restrict__ W, int K,
    const float* __restrict__ bias,
    const float* __restrict__ residual,   // nullptr -> skip; ld = 1024
    float* __restrict__ out, int ldo)
{
    __shared__ float As[2 * TILE_FLOATS];   // double-buffered A tile
    __shared__ float Bs[2 * TILE_FLOATS];   // double-buffered W tile

    const int tid     = threadIdx.x;
    const int lane    = tid & 31;
    const int wave    = tid >> 5;          // 0..7
    const int lrow    = lane & 15;         // M (A frag) / N (B frag) in 16
    const int khalf   = (lane >> 4) << 1;  // K offset 0 or 2 within k-step
    const int rowHalf = (lane >> 4) << 3;  // C/D: M offset 0 or 8

    const int mWave = (wave & 3) * 32;     // 4 waves tile M
    const int nWave = (wave >> 2) * 64;    // 2 waves tile N

    const int blockN = blockIdx.x * 128;
    const int blockM = blockIdx.y * 128;

    // Per-thread async-copy slots: 4 x b128 for A, 4 x b128 for W per chunk.
    int aoff[4], boff[4];                  // element offsets within tile pane
    unsigned ldsA0[4], ldsB0[4];           // LDS byte addresses (buffer 0)
#pragma unroll
    for (int i = 0; i < 4; ++i) {
        int idx = tid + i * 256;           // 0..1023 float4 slots
        int r   = idx >> 3;                // 128 rows, 8 float4 per row
        int c   = (idx & 7) << 2;          // float column within chunk
        aoff[i]  = r * lda + c;
        boff[i]  = r * K + c;
        ldsA0[i] = (unsigned)(size_t)&As[r * LDSS + c];
        ldsB0[i] = (unsigned)(size_t)&Bs[r * LDSS + c];
    }
    const float* Abase = A + (size_t)blockM * lda;
    const float* Wbase = W + (size_t)blockN * K;

    v8f acc[2][4];
#pragma unroll
    for (int mt = 0; mt < 2; ++mt)
#pragma unroll
        for (int nt = 0; nt < 4; ++nt)
            acc[mt][nt] = v8f{0.f, 0.f, 0.f, 0.f, 0.f, 0.f, 0.f, 0.f};

    // Prologue: chunk 0 -> buffer 0
#pragma unroll
    for (int i = 0; i < 4; ++i) {
        async_b128_to_lds(ldsA0[i], aoff[i] * 4, Abase);
        async_b128_to_lds(ldsB0[i], boff[i] * 4, Wbase);
    }

    const int nchunks = K >> 5;            // K / 32
    for (int ci = 0; ci < nchunks; ++ci) {
        if (ci + 1 < nchunks) {
            // Issue next chunk into the other buffer, overlapping compute.
            const unsigned nbuf = (unsigned)((ci + 1) & 1) * TILE_BYTES;
            const int kb4 = (ci + 1) << 7;           // 32 floats = 128 bytes
#pragma unroll
            for (int i = 0; i < 4; ++i) {
                async_b128_to_lds(ldsA0[i] + nbuf, aoff[i] * 4 + kb4, Abase);
                async_b128_to_lds(ldsB0[i] + nbuf, boff[i] * 4 + kb4, Wbase);
            }
            asm volatile("s_wait_asynccnt 0x8" ::: "memory");  // chunk ci done
        } else {
            asm volatile("s_wait_asynccnt 0x0" ::: "memory");
        }
        __syncthreads();                   // all waves' chunk-ci data landed

        const float* AsBuf = As + (ci & 1) * TILE_FLOATS;
        const float* BsBuf = Bs + (ci & 1) * TILE_FLOATS;

#pragma unroll
        for (int kk = 0; kk < 32; kk += 4) {
            v2f a[2], b[4];
#pragma unroll
            for (int mt = 0; mt < 2; ++mt)
                a[mt] = *(const v2f*)&AsBuf[(mWave + mt * 16 + lrow) * LDSS + kk + khalf];
#pragma unroll
            for (int nt = 0; nt < 4; ++nt)
                b[nt] = *(const v2f*)&BsBuf[(nWave + nt * 16 + lrow) * LDSS + kk + khalf];
#pragma unroll
            for (int mt = 0; mt < 2; ++mt)
#pragma unroll
                for (int nt = 0; nt < 4; ++nt)
                    acc[mt][nt] = __builtin_amdgcn_wmma_f32_16x16x4_f32(
                        false, a[mt], false, b[nt],
                        (short)0, acc[mt][nt], false, false);
        }
        __syncthreads();                   // readers done; buffer reusable
    }

    // Epilogue: C/D layout -> lane holds N = lrow, rows M = rowHalf + r
#pragma unroll
    for (int mt = 0; mt < 2; ++mt) {
#pragma unroll
        for (int nt = 0; nt < 4; ++nt) {
            int gn = blockN + nWave + nt * 16 + lrow;
            float bv = bias[gn];
#pragma unroll
            for (int r = 0; r < 8; ++r) {
                int gm = blockM + mWave + mt * 16 + rowHalf + r;
                float v = acc[mt][nt][r] + bv;
                if (residual) v += residual[(size_t)gm * 1024 + gn];
                out[(size_t)gm * ldo + gn] = v;
            }
        }
    }
}

// y = LN(x_row) * g + beta, written at out[row*2048 + colOffset + col]
__global__ __launch_bounds__(256) void ln_kernel(
    const float* __restrict__ x,      // [B,1024]
    const float* __restrict__ g,
    const float* __restrict__ beta,
    float* __restrict__ out,          // [B,2048]
    int colOffset)
{
    const int row = blockIdx.x;
    const int t   = threadIdx.x;

    float4 v = *(const float4*)&x[(size_t)row * 1024 + t * 4];
    float s  = v.x + v.y + v.z + v.w;
    float ss = v.x * v.x + v.y * v.y + v.z * v.z + v.w * v.w;

#pragma unroll
    for (int o = 16; o > 0; o >>= 1) {     // wave32 reduce
        s  += __shfl_xor(s, o);
        ss += __shfl_xor(ss, o);
    }
    __shared__ float sbuf[8], ssbuf[8];
    if ((t & 31) == 0) { sbuf[t >> 5] = s; ssbuf[t >> 5] = ss; }
    __syncthreads();
    float tot = 0.f, tot2 = 0.f;
#pragma unroll
    for (int w = 0; w < 8; ++w) { tot += sbuf[w]; tot2 += ssbuf[w]; }

    const float mu  = tot * (1.0f / 1024.0f);
    const float var = tot2 * (1.0f / 1024.0f) - mu * mu;
    const float inv = rsqrtf(var + 1e-5f);

    float4 gv = *(const float4*)&g[t * 4];
    float4 bv = *(const float4*)&beta[t * 4];
    float4 o;
    o.x = (v.x - mu) * inv * gv.x + bv.x;
    o.y = (v.y - mu) * inv * gv.y + bv.y;
    o.z = (v.z - mu) * inv * gv.z + bv.z;
    o.w = (v.w - mu) * inv * gv.w + bv.w;
    *(float4*)&out[(size_t)row * 2048 + colOffset + t * 4] = o;
}

extern "C" void kernel_launch(void* const* d_in, const int* in_sizes, int n_in,
                              void* d_out, int out_size, void* d_ws, size_t ws_size,
                              hipStream_t stream) {
    (void)in_sizes; (void)n_in; (void)out_size; (void)ws_size;

    const float* eeg   = (const float*)d_in[0];
    const float* ecg   = (const float*)d_in[1];
    const float* Wv1   = (const float*)d_in[2];
    const float* bv1   = (const float*)d_in[3];
    const float* Wo1   = (const float*)d_in[4];
    const float* bo1   = (const float*)d_in[5];
    const float* Wv2   = (const float*)d_in[6];
    const float* bv2   = (const float*)d_in[7];
    const float* Wo2   = (const float*)d_in[8];
    const float* bo2   = (const float*)d_in[9];
    const float* g1    = (const float*)d_in[10];
    const float* beta1 = (const float*)d_in[11];
    const float* g2    = (const float*)d_in[12];
    const float* beta2 = (const float*)d_in[13];
    const float* Wf    = (const float*)d_in[14];
    const float* bf    = (const float*)d_in[15];
    float* out = (float*)d_out;

    const size_t B = 32768, D = 1024;
    float* fused = (float*)d_ws;          // [B, 2D] (256 MB)
    float* T1    = fused;                 // [B, D]  (reuses fused 1st half)
    float* T2    = fused + B * D;         // [B, D]  (reuses fused 2nd half)
    float* R1    = fused + B * 2 * D;     // [B, D]
    float* R2    = R1 + B * D;            // [B, D]

    dim3 block(256);
    dim3 grid(1024 / 128, 32768 / 128);   // (N tiles, M tiles)

    // T1 = ecg @ Wv1^T + bv1 ; T2 = eeg @ Wv2^T + bv2
    gemm_f32_wmma<<<grid, block, 0, stream>>>(ecg, 1024, Wv1, 1024, bv1, nullptr, T1, 1024);
    gemm_f32_wmma<<<grid, block, 0, stream>>>(eeg, 1024, Wv2, 1024, bv2, nullptr, T2, 1024);
    // R1 = eeg + T1 @ Wo1^T + bo1 ; R2 = ecg + T2 @ Wo2^T + bo2
    gemm_f32_wmma<<<grid, block, 0, stream>>>(T1, 1024, Wo1, 1024, bo1, eeg, R1, 1024);
    gemm_f32_wmma<<<grid, block, 0, stream>>>(T2, 1024, Wo2, 1024, bo2, ecg, R2, 1024);
    // fused[:, :D] = LN(R1) ; fused[:, D:] = LN(R2)   (T1/T2 are dead now)
    ln_kernel<<<32768, block, 0, stream>>>(R1, g1, beta1, fused, 0);
    ln_kernel<<<32768, block, 0, stream>>>(R2, g2, beta2, fused, 1024);
    // out = fused @ Wf^T + bf   (K = 2048)
    gemm_f32_wmma<<<grid, block, 0, stream>>>(fused, 2048, Wf, 2048, bf, nullptr, out, 1024);
}